// YoloNASPoseTaskAlignedAssigner_90529320665204
// MI455X (gfx1250) — compile-verified
//
#include <hip/hip_runtime.h>
#include <hip/hip_bf16.h>

#define TOPK 13
#define EPSF 1e-9f
#define MAXL 8448   // >= L (8400), padded
#define MAXN 64     // >= N (64); claim mask is one u64 per anchor

typedef __attribute__((ext_vector_type(2))) float v2f;
typedef __attribute__((ext_vector_type(8))) float v8f;

// ---------------------------------------------------------------- k0: init
__global__ __launch_bounds__(256) void k0_init(int* cnt, unsigned long long* bits,
                                               unsigned int* maxmet, unsigned int* maxiou,
                                               int BL, int BN) {
  int i = blockIdx.x * 256 + threadIdx.x;
  if (i < BL) { cnt[i] = 0; bits[i] = 0ull; }
  if (i < BN) { maxmet[i] = 0u; maxiou[i] = 0u; }
}

__device__ __forceinline__ float iou_f(float4 p, float4 g, float areaP, float areaG) {
  float lx = fmaxf(p.x, g.x), ly = fmaxf(p.y, g.y);
  float rx = fminf(p.z, g.z), ry = fminf(p.w, g.w);
  float w = fmaxf(rx - lx, 0.0f), h = fmaxf(ry - ly, 0.0f);
  float inter = w * h;
  return inter / (areaP + areaG - inter + EPSF);
}

// --------------------------------------------- k1: per-(b,n) metric + top-13
__global__ __launch_bounds__(256) void k1_topk(const float* __restrict__ scores,
                                               const float* __restrict__ bboxes,
                                               const float* __restrict__ anchors,
                                               const int* __restrict__ glabels,
                                               const float* __restrict__ gboxes,
                                               const float* __restrict__ pad,
                                               int* __restrict__ cnt,
                                               unsigned long long* __restrict__ bits,
                                               int L, int C, int N) {
  int bn = blockIdx.x;
  int b = bn / N, n = bn % N;
  if (pad[bn] <= 0.5f) return;  // padded gt: contributes nothing (uniform exit)

  __shared__ float sM[MAXL];
  __shared__ unsigned char sF[MAXL];
  __shared__ float rV[256];
  __shared__ int   rI[256];

  const float4 g = ((const float4*)gboxes)[bn];
  const float areaG = (g.z - g.x) * (g.w - g.y);
  const int lab = glabels[bn];
  const int t = threadIdx.x;

  for (int l = t; l < L; l += 256) {
    float4 p = ((const float4*)bboxes)[(size_t)b * L + l];
    __builtin_prefetch(&bboxes[((size_t)b * L + l + 256) * 4], 0, 1);
    float areaP = (p.z - p.x) * (p.w - p.y);
    float iou = iou_f(p, g, areaP, areaG);
    float ax = anchors[2 * l], ay = anchors[2 * l + 1];
    float md = fminf(fminf(ax - g.x, ay - g.y), fminf(g.z - ax, g.w - ay));
    bool ing = md > EPSF;
    float sc = scores[((size_t)b * L + l) * C + lab];
    float i2 = iou * iou;
    float met = sc * (i2 * i2 * i2);          // score^ALPHA * iou^BETA, ALPHA=1 BETA=6
    sM[l] = ing ? met : 0.0f;                 // topk input is metric * is_in_gts
    sF[l] = ing ? 1 : 0;
  }
  __syncthreads();

  // 13 stable argmax passes (ties -> lowest index, matching jax.lax.top_k)
  for (int k = 0; k < TOPK; ++k) {
    float bv = -1.0e30f; int bi = 0x7fffffff;
    for (int l = t; l < L; l += 256) {        // within a thread, first hit = lowest l
      float v = sM[l];
      if (v > bv) { bv = v; bi = l; }
    }
    rV[t] = bv; rI[t] = bi;
    __syncthreads();
    for (int s = 128; s > 0; s >>= 1) {
      if (t < s) {
        float v = rV[t + s]; int i = rI[t + s];
        if (v > rV[t] || (v == rV[t] && i < rI[t])) { rV[t] = v; rI[t] = i; }
      }
      __syncthreads();
    }
    if (t == 0) {
      int idx = rI[0];
      if (idx < L) {
        if (sF[idx]) {                        // counts only if is_in_gts
          atomicAdd(&cnt[(size_t)b * L + idx], 1);
          atomicOr(&bits[(size_t)b * L + idx], 1ull << n);
        }
        sM[idx] = -2.0e30f;                   // exclude from later passes
      }
    }
    __syncthreads();
  }
}

// ------------------------------- k2: per-(b,l) resolve assignment + normalizers
__global__ __launch_bounds__(256) void k2_assign(const float* __restrict__ scores,
                                                 const float* __restrict__ bboxes,
                                                 const int* __restrict__ glabels,
                                                 const float* __restrict__ gboxes,
                                                 const int* __restrict__ cnt,
                                                 const unsigned long long* __restrict__ bits,
                                                 int* __restrict__ an,
                                                 float* __restrict__ amet,
                                                 float* __restrict__ aiou,
                                                 unsigned int* __restrict__ maxmet,
                                                 unsigned int* __restrict__ maxiou,
                                                 int L, int C, int N) {
  __shared__ float4 sG[MAXN];
  int b = blockIdx.y;
  int t = threadIdx.x;
  if (t < N) sG[t] = ((const float4*)gboxes)[(size_t)b * N + t];
  __syncthreads();

  int l = blockIdx.x * 256 + t;
  if (l >= L) return;

  float4 p = ((const float4*)bboxes)[(size_t)b * L + l];
  float areaP = (p.z - p.x) * (p.w - p.y);

  float bestIou = -1.0f; int nBest = 0;       // argmax over n, first max wins
  for (int n = 0; n < N; ++n) {
    float4 g = sG[n];
    float areaG = (g.z - g.x) * (g.w - g.y);
    float v = iou_f(p, g, areaP, areaG);
    if (v > bestIou) { bestIou = v; nBest = n; }
  }

  int c = cnt[(size_t)b * L + l];
  unsigned long long m = bits[(size_t)b * L + l];
  int ns;
  if (c > 1)       ns = nBest;                // column replaced by is_max_iou
  else if (c == 1) ns = __ffsll(m) - 1;       // unique claimant
  else             ns = -1;                   // background

  an[(size_t)b * L + l] = ns;
  float mv = 0.0f, iv = 0.0f;
  if (ns >= 0) {
    float4 g = sG[ns];
    float areaG = (g.z - g.x) * (g.w - g.y);
    iv = iou_f(p, g, areaP, areaG);
    int lab = glabels[(size_t)b * N + ns];
    float sc = scores[((size_t)b * L + l) * C + lab];
    float i2 = iv * iv;
    mv = sc * (i2 * i2 * i2);
    atomicMax(&maxmet[(size_t)b * N + ns], __float_as_uint(mv));  // nonneg floats
    atomicMax(&maxiou[(size_t)b * N + ns], __float_as_uint(iv));
  }
  amet[(size_t)b * L + l] = mv;
  aiou[(size_t)b * L + l] = iv;
}

// --------------------------- k3: labels / bboxes / scores / gt_index outputs
__global__ __launch_bounds__(256) void k3_out(const int* __restrict__ glabels,
                                              const float* __restrict__ gboxes,
                                              const int* __restrict__ an,
                                              const float* __restrict__ amet,
                                              const unsigned int* __restrict__ maxmet,
                                              const unsigned int* __restrict__ maxiou,
                                              const int* __restrict__ bgp,
                                              float* __restrict__ outLab,
                                              float* __restrict__ outBox,
                                              float* __restrict__ outScore,
                                              float* __restrict__ outIdx,
                                              int L, int C, int N) {
  int b = blockIdx.y;
  int l = blockIdx.x * 256 + threadIdx.x;
  if (l >= L) return;
  size_t bl = (size_t)b * L + l;
  int ns = an[bl];
  int g = ns < 0 ? 0 : ns;                    // argmax of all-zero mask -> 0
  int bg = *bgp;
  int lab = glabels[(size_t)b * N + g];
  int labOut = (ns >= 0) ? lab : bg;
  outLab[bl] = (float)labOut;
  ((float4*)outBox)[bl] = ((const float4*)gboxes)[(size_t)b * N + g];
  float am = 0.0f;
  if (ns >= 0) {
    float mm = __uint_as_float(maxmet[(size_t)b * N + ns]);
    float mi = __uint_as_float(maxiou[(size_t)b * N + ns]);
    am = amet[bl] / (mm + EPSF) * mi;
  }
  for (int cc = 0; cc < C; ++cc)
    outScore[bl * C + cc] = (ns >= 0 && lab == cc) ? am : 0.0f;
  outIdx[bl] = (float)(g + b * N);
}

// ------------- k4: pose gather as one-hot GEMM via V_WMMA_F32_16X16X4_F32
// assigned_poses[b] (L x 51) = OneHot(L x 64) * gt_poses[b] (64 x 51)
// Per block: one batch b, 128 anchors (8 waves x 16 rows), 4 col-tiles of 16.
// B matrix staged in LDS K-major with padded stride 68 so each lane's (k0,k0+1)
// pair is one aligned float2 LDS load, bank-conflict-free wave-wide.
#define BSTR 68
__global__ __launch_bounds__(256) void k4_pose(const float* __restrict__ gposes,
                                               const int* __restrict__ an,
                                               float* __restrict__ outPose,
                                               int L, int N, int K3) {
  __shared__ __align__(16) float sBT[64 * BSTR];  // [col][k], k-padded stride 68
  __shared__ int sIdx[128];
  int b = blockIdx.y;
  int t = threadIdx.x;
  int lbase = blockIdx.x * 128;

  for (int e = t; e < 64 * 64; e += 256) {
    int row = e >> 6, col = e & 63;           // row = gt index (K), col = pose dim (N)
    float v = 0.0f;
    if (row < N && col < K3) v = gposes[((size_t)b * N + row) * K3 + col];
    sBT[col * BSTR + row] = v;                // transposed, padded
  }
  if (t < 128) {
    int l = lbase + t;
    int v = 0;
    if (l < L) { int ns = an[(size_t)b * L + l]; v = ns < 0 ? 0 : ns; }
    sIdx[t] = v;
  }
  __syncthreads();

  int wave = t >> 5, lane = t & 31;           // wave32
  int m16 = lane & 15;                        // A-matrix row M / B,C,D column N
  int half = (lane < 16) ? 0 : 2;             // K sub-offset per half-wave
  int idx = sIdx[wave * 16 + m16];            // gather index for row M=m16

  for (int tile = 0; tile < 4; ++tile) {
    v8f acc = {};
    int colBase = (tile * 16 + m16) * BSTR + half;   // &sBT[col][half]
    const v2f* bcol = (const v2f*)&sBT[colBase];     // k0 even -> 8B aligned
#pragma unroll
    for (int c = 0; c < 16; ++c) {            // 16 K-chunks of 4 -> K total 64
      int k0 = c * 4 + half;
      v2f a, bm;
      a.x = (idx == k0)     ? 1.0f : 0.0f;    // one-hot A row
      a.y = (idx == k0 + 1) ? 1.0f : 0.0f;
      bm = bcol[c * 2];                       // {sBT[col][k0], sBT[col][k0+1]}
      // D = A(16x4) * B(4x16) + C ; 8-arg f32 WMMA
      acc = __builtin_amdgcn_wmma_f32_16x16x4_f32(false, a, false, bm,
                                                  (short)0, acc, false, false);
    }
    int col = tile * 16 + m16;
    if (col < K3) {
#pragma unroll
      for (int j = 0; j < 8; ++j) {
        int m = j + ((lane < 16) ? 0 : 8);    // C/D layout: vgpr j -> M=j / j+8
        int l = lbase + wave * 16 + m;
        if (l < L) outPose[((size_t)b * L + l) * K3 + col] = acc[j];
      }
    }
  }
}

// ---------------------------------------------------------------- launcher
extern "C" void kernel_launch(void* const* d_in, const int* in_sizes, int n_in,
                              void* d_out, int out_size, void* d_ws, size_t ws_size,
                              hipStream_t stream) {
  const float* pred_scores  = (const float*)d_in[0];   // (B,L,C)
  const float* pred_bboxes  = (const float*)d_in[1];   // (B,L,4)
  // d_in[2] = pred_poses (unused by reference outputs except K)
  const float* anchor_pts   = (const float*)d_in[3];   // (L,2)
  const int*   gt_labels    = (const int*)d_in[4];     // (B,N,1)
  const float* gt_bboxes    = (const float*)d_in[5];   // (B,N,4)
  const float* gt_poses     = (const float*)d_in[6];   // (B,N,K,3)
  const float* pad_gt_mask  = (const float*)d_in[7];   // (B,N,1)
  const int*   bg_index_p   = (const int*)d_in[8];     // scalar

  int L  = in_sizes[3] / 2;
  int B  = in_sizes[1] / (L * 4);
  int C  = in_sizes[0] / (B * L);
  int N  = in_sizes[4] / B;
  int K3 = in_sizes[2] / (B * L);              // K*3 = 51

  size_t BL = (size_t)B * L, BN = (size_t)B * N;

  // workspace carve-up (8B-aligned first)
  char* w = (char*)d_ws;
  unsigned long long* bits = (unsigned long long*)w; w += BL * 8;
  int*          cnt    = (int*)w;          w += BL * 4;
  int*          an     = (int*)w;          w += BL * 4;
  float*        amet   = (float*)w;        w += BL * 4;
  float*        aiou   = (float*)w;        w += BL * 4;
  unsigned int* maxmet = (unsigned int*)w; w += BN * 4;
  unsigned int* maxiou = (unsigned int*)w; w += BN * 4;
  (void)aiou; (void)ws_size;

  float* out      = (float*)d_out;
  float* outLab   = out;
  float* outBox   = outLab + BL;
  float* outPose  = outBox + BL * 4;
  float* outScore = outPose + BL * K3;
  float* outIdx   = outScore + BL * C;

  dim3 blk(256);
  int initBlocks = (int)((BL + 255) / 256);
  k0_init<<<initBlocks, blk, 0, stream>>>(cnt, bits, maxmet, maxiou, (int)BL, (int)BN);

  k1_topk<<<B * N, blk, 0, stream>>>(pred_scores, pred_bboxes, anchor_pts,
                                     gt_labels, gt_bboxes, pad_gt_mask,
                                     cnt, bits, L, C, N);

  dim3 g2((L + 255) / 256, B);
  k2_assign<<<g2, blk, 0, stream>>>(pred_scores, pred_bboxes, gt_labels, gt_bboxes,
                                    cnt, bits, an, amet, aiou, maxmet, maxiou,
                                    L, C, N);

  k3_out<<<g2, blk, 0, stream>>>(gt_labels, gt_bboxes, an, amet, maxmet, maxiou,
                                 bg_index_p, outLab, outBox, outScore, outIdx,
                                 L, C, N);

  dim3 g4((L + 127) / 128, B);
  k4_pose<<<g4, blk, 0, stream>>>(gt_poses, an, outPose, L, N, K3);
}